// DGMC_67912022884658
// MI455X (gfx1250) — compile-verified
//
#include <hip/hip_runtime.h>
#include <math.h>

// ---------------------------------------------------------------------------
// DGMC pipeline for MI455X (gfx1250, wave32).
// GEMMs use native fp32 WMMA (V_WMMA_F32_16X16X4_F32): workload is
// memory-bound on edge gather/scatter (~2.5 GB/call vs 23.3 TB/s), so fp32
// matrix math is free and keeps reference precision.
// ---------------------------------------------------------------------------

#define D      128
#define NLAY   3
#define DBIL   64
#define DMLP   192   // DBIL * NLAY
#define NQn    1000
#define NTn    100000
#define EQn    16000
#define ETn    800000
#define KVn    512
#define NMAPn  200

typedef __attribute__((ext_vector_type(2))) float v2f;
typedef __attribute__((ext_vector_type(8))) float v8f;

// ---------------------------------------------------------------- utilities
__global__ void fill_zero(float* __restrict__ p, long long n) {
    long long i = (long long)blockIdx.x * blockDim.x + threadIdx.x;
    if (i < n) p[i] = 0.0f;
}

__global__ void deg_count(const int* __restrict__ dst, int E, float* __restrict__ deg) {
    int e = blockIdx.x * blockDim.x + threadIdx.x;
    if (e < E) atomicAdd(&deg[dst[e]], 1.0f);
}

__global__ void make_dinv(float* __restrict__ dinv, int N) {
    int i = blockIdx.x * blockDim.x + threadIdx.x;
    if (i < N) dinv[i] = rsqrtf(dinv[i] + 1.0f);   // deg + self-loop
}

// ------------------------------------------------------------- WMMA GEMM
// H[N,128] = X[N,128] @ W[128,128], row-major, fp32 WMMA 16x16x4.
// One wave per 16x16 output tile; 8 waves (256 threads) per block.
// A layout (16x4 f32): lanes 0-15: M=lane, K={k,k+1}; lanes 16-31: M=lane-16, K={k+2,k+3}
// B layout (4x16 f32): mirrors A striping (row per half-wave, N=lane&15).
// D layout (16x16 f32): VGPR r = row (r + 8*half), col = lane&15.
__global__ void gemm_xw_wmma(const float* __restrict__ X, const float* __restrict__ W,
                             float* __restrict__ H, int N, int tilesM) {
    const int wave = threadIdx.x >> 5;
    const int tile = blockIdx.x * 8 + wave;
    const int totalTiles = tilesM * (D / 16);
    if (tile >= totalTiles) return;                 // wave-uniform exit (EXEC stays full)

    const int tm = tile >> 3;                       // D/16 == 8 column tiles
    const int tn = tile & 7;
    const int m0 = tm * 16, n0 = tn * 16;
    const int lane = threadIdx.x & 31;
    const int hf  = lane >> 4;                      // 0: K lo pair, 1: K hi pair
    const int l16 = lane & 15;

    int arow = m0 + l16;
    if (arow >= N) arow = N - 1;                    // tail clamp (stores are guarded)
    const float* __restrict__ Xr = X + (long long)arow * D;

    v8f acc = {};
    #pragma unroll
    for (int k = 0; k < D; k += 4) {
        const int ka = k + 2 * hf;
        v2f a; a.x = Xr[ka];                 a.y = Xr[ka + 1];
        v2f b; b.x = W[ka * D + n0 + l16];   b.y = W[(ka + 1) * D + n0 + l16];
        acc = __builtin_amdgcn_wmma_f32_16x16x4_f32(
                  /*neg_a=*/false, a, /*neg_b=*/false, b,
                  /*c_mod=*/(short)0, acc, /*reuse_a=*/false, /*reuse_b=*/false);
    }
    #pragma unroll
    for (int r = 0; r < 8; ++r) {
        const int row = m0 + r + 8 * hf;
        if (row < N) H[(long long)row * D + n0 + l16] = acc[r];
    }
}

// -------------------------------------------------- edge message scatter-add
// 32 lanes per edge, float4 per lane: Agg[dst] += h[src] * dinv[src]*dinv[dst]
__global__ void gcn_scatter(const float* __restrict__ H, const int* __restrict__ src,
                            const int* __restrict__ dst, const float* __restrict__ dinv,
                            float* __restrict__ Agg, int E) {
    const long long gid = (long long)blockIdx.x * blockDim.x + threadIdx.x;
    const long long e = gid >> 5;
    if (e >= E) return;
    const int f = (int)(gid & 31) * 4;
    const int s = src[e], d = dst[e];
    const float w = dinv[s] * dinv[d];
    const float4 h = *reinterpret_cast<const float4*>(H + (long long)s * D + f);
    float* ap = Agg + (long long)d * D + f;
    atomicAdd(ap + 0, h.x * w);
    atomicAdd(ap + 1, h.y * w);
    atomicAdd(ap + 2, h.z * w);
    atomicAdd(ap + 3, h.w * w);
}

// ------------------------------------- self-loop + bias + l2norm (+ELU) -> X
// One wave per row; wave32 shuffle reduction for the norm.
__global__ void gcn_finalize(const float* __restrict__ Agg, const float* __restrict__ H,
                             const float* __restrict__ dinv, const float* __restrict__ bias,
                             float* __restrict__ Xout, int N, int apply_elu) {
    const int lane = threadIdx.x & 31;
    const int wid  = threadIdx.x >> 5;
    const int row  = blockIdx.x * 8 + wid;
    if (row >= N) return;
    const float di = dinv[row];
    const float sl = di * di;
    const long long base = (long long)row * D + lane * 4;
    const float4 a = *reinterpret_cast<const float4*>(Agg + base);
    const float4 h = *reinterpret_cast<const float4*>(H + base);
    const float4 b = *reinterpret_cast<const float4*>(bias + lane * 4);
    float4 v;
    v.x = a.x + sl * h.x + b.x;
    v.y = a.y + sl * h.y + b.y;
    v.z = a.z + sl * h.z + b.z;
    v.w = a.w + sl * h.w + b.w;
    float ss = v.x * v.x + v.y * v.y + v.z * v.z + v.w * v.w;
    #pragma unroll
    for (int o = 16; o > 0; o >>= 1) ss += __shfl_xor(ss, o, 32);
    const float inv = 1.0f / fmaxf(sqrtf(ss), 1e-12f);
    v.x *= inv; v.y *= inv; v.z *= inv; v.w *= inv;
    if (apply_elu) {
        v.x = v.x > 0.0f ? v.x : expf(v.x) - 1.0f;
        v.y = v.y > 0.0f ? v.y : expf(v.y) - 1.0f;
        v.z = v.z > 0.0f ? v.z : expf(v.z) - 1.0f;
        v.w = v.w > 0.0f ? v.w : expf(v.w) - 1.0f;
    }
    *reinterpret_cast<float4*>(Xout + base) = v;
}

// ----------------------------------------------- consensus: Xt[v_map]=Xq[u_map]
__global__ void consensus(float* __restrict__ Xt, const float* __restrict__ Xq,
                          const int* __restrict__ u_map, const int* __restrict__ v_map,
                          int nmap) {
    const int i = blockIdx.x;
    const int f = threadIdx.x;   // 128 threads
    if (i < nmap)
        Xt[(long long)v_map[i] * D + f] = Xq[(long long)u_map[i] * D + f];
}

// --------------------------------- bilinear stage 1: T[o,k] = sum_j q[j]*W[o,j,k]
__global__ void bil_qW(const float* __restrict__ Xq, const int* __restrict__ u_ptr,
                       const float* __restrict__ Wb, float* __restrict__ T) {
    const int o = blockIdx.x;    // 64
    const int k = threadIdx.x;   // 128
    const int u = *u_ptr;
    const float* __restrict__ q = Xq + (long long)u * D;
    const float* __restrict__ w = Wb + (long long)o * D * D + k;
    float acc = 0.0f;
    #pragma unroll 8
    for (int j = 0; j < D; ++j) acc += q[j] * w[(long long)j * D];
    T[o * D + k] = acc;
}

// ------------------- bilinear stage 2: feat[v, layer*64+o] = <Xt[v_li[v]], T[o]> + b
__global__ void bil_out(const float* __restrict__ Xt, const int* __restrict__ v_li,
                        const float* __restrict__ T, const float* __restrict__ bb,
                        float* __restrict__ feat, int layer) {
    const int v = blockIdx.x;    // 512
    const int o = threadIdx.x;   // 64
    const float* __restrict__ xr = Xt + (long long)v_li[v] * D;
    const float* __restrict__ tr = T + o * D;
    float acc = 0.0f;
    #pragma unroll 8
    for (int k = 0; k < D; ++k) acc += xr[k] * tr[k];
    feat[(long long)v * DMLP + layer * DBIL + o] = acc + bb[o];
}

// --------------------------------------------------------------- tiny MLP head
__global__ void mlp_head(const float* __restrict__ feat,
                         const float* __restrict__ W1, const float* __restrict__ b1,
                         const float* __restrict__ W2, const float* __restrict__ b2,
                         const float* __restrict__ W3, const float* __restrict__ b3,
                         float* __restrict__ out) {
    __shared__ float sf[DMLP];
    __shared__ float s1[96];
    __shared__ float s2[48];
    const int v = blockIdx.x;    // 512 blocks
    const int t = threadIdx.x;   // 128 threads
    for (int i = t; i < DMLP; i += 128) sf[i] = feat[(long long)v * DMLP + i];
    __syncthreads();
    for (int c = t; c < 96; c += 128) {
        float a = b1[c];
        for (int r = 0; r < DMLP; ++r) a += sf[r] * W1[r * 96 + c];
        s1[c] = fmaxf(a, 0.0f);
    }
    __syncthreads();
    for (int c = t; c < 48; c += 128) {
        float a = b2[c];
        for (int r = 0; r < 96; ++r) a += s1[r] * W2[r * 48 + c];
        s2[c] = fmaxf(a, 0.0f);
    }
    __syncthreads();
    if (t == 0) {
        float a = b3[0];
        for (int r = 0; r < 48; ++r) a += s2[r] * W3[r];
        out[v] = a;
    }
}

// ===========================================================================
extern "C" void kernel_launch(void* const* d_in, const int* in_sizes, int n_in,
                              void* d_out, int out_size, void* d_ws, size_t ws_size,
                              hipStream_t stream) {
    (void)in_sizes; (void)n_in; (void)out_size; (void)ws_size;

    const float* x_q   = (const float*)d_in[0];
    const float* x_t   = (const float*)d_in[1];
    const int*   eq    = (const int*)d_in[2];
    const int*   et    = (const int*)d_in[3];
    const int*   u_ptr = (const int*)d_in[4];
    const int*   v_li  = (const int*)d_in[5];
    const int*   u_map = (const int*)d_in[6];
    const int*   v_map = (const int*)d_in[7];
    const float* W_gnn = (const float*)d_in[8];
    const float* b_gnn = (const float*)d_in[9];
    const float* W_bil = (const float*)d_in[10];
    const float* b_bil = (const float*)d_in[11];
    const float* W1    = (const float*)d_in[12];
    const float* b1    = (const float*)d_in[13];
    const float* W2    = (const float*)d_in[14];
    const float* b2    = (const float*)d_in[15];
    const float* W3    = (const float*)d_in[16];
    const float* b3    = (const float*)d_in[17];

    const int* sq = eq,  *dq = eq + EQn;
    const int* st = et,  *dt = et + ETn;

    // ---- workspace carve (~156 MB of floats) ----
    float* p = (float*)d_ws;
    float* Xt     = p; p += (long long)NTn * D;
    float* Ht     = p; p += (long long)NTn * D;
    float* AggT   = p; p += (long long)NTn * D;
    float* Xq     = p; p += (long long)NQn * D;
    float* Hq     = p; p += (long long)NQn * D;
    float* AggQ   = p; p += (long long)NQn * D;
    float* dinv_t = p; p += NTn;
    float* dinv_q = p; p += NQn;
    float* featb  = p; p += (long long)KVn * DMLP;
    float* Tmat   = p; p += DBIL * D;

    float* out    = (float*)d_out;            // [512]
    float* out_Xq = out + KVn;                // [1000*128]
    float* out_Xt = out_Xq + (long long)NQn * D;   // [100000*128]

    // working copies (inputs must stay untouched)
    hipMemcpyAsync(Xq, x_q, (size_t)NQn * D * sizeof(float), hipMemcpyDeviceToDevice, stream);
    hipMemcpyAsync(Xt, x_t, (size_t)NTn * D * sizeof(float), hipMemcpyDeviceToDevice, stream);

    // ---- degree / dinv (edge-only -> constant across layers) ----
    fill_zero<<<(NTn + 255) / 256, 256, 0, stream>>>(dinv_t, NTn);
    fill_zero<<<(NQn + 255) / 256, 256, 0, stream>>>(dinv_q, NQn);
    deg_count<<<(ETn + 255) / 256, 256, 0, stream>>>(dt, ETn, dinv_t);
    deg_count<<<(EQn + 255) / 256, 256, 0, stream>>>(dq, EQn, dinv_q);
    make_dinv<<<(NTn + 255) / 256, 256, 0, stream>>>(dinv_t, NTn);
    make_dinv<<<(NQn + 255) / 256, 256, 0, stream>>>(dinv_q, NQn);

    const int tilesQ = (NQn + 15) / 16;   // 63
    const int tilesT = (NTn + 15) / 16;   // 6250
    const long long aggT_elems = (long long)NTn * D;
    const long long aggQ_elems = (long long)NQn * D;

    for (int i = 0; i < NLAY; ++i) {
        const float* Wl = W_gnn + (long long)i * D * D;
        const float* bl = b_gnn + i * D;
        const int elu = (i < NLAY - 1) ? 1 : 0;

        // ---- query graph ----
        gemm_xw_wmma<<<(tilesQ * 8 + 7) / 8, 256, 0, stream>>>(Xq, Wl, Hq, NQn, tilesQ);
        fill_zero<<<(int)((aggQ_elems + 255) / 256), 256, 0, stream>>>(AggQ, aggQ_elems);
        gcn_scatter<<<(int)(((long long)EQn * 32 + 255) / 256), 256, 0, stream>>>(
            Hq, sq, dq, dinv_q, AggQ, EQn);
        gcn_finalize<<<(NQn + 7) / 8, 256, 0, stream>>>(AggQ, Hq, dinv_q, bl, Xq, NQn, elu);

        // ---- target graph ----
        gemm_xw_wmma<<<(tilesT * 8 + 7) / 8, 256, 0, stream>>>(Xt, Wl, Ht, NTn, tilesT);
        fill_zero<<<(int)((aggT_elems + 255) / 256), 256, 0, stream>>>(AggT, aggT_elems);
        gcn_scatter<<<(int)(((long long)ETn * 32 + 255) / 256), 256, 0, stream>>>(
            Ht, st, dt, dinv_t, AggT, ETn);
        gcn_finalize<<<(NTn + 7) / 8, 256, 0, stream>>>(AggT, Ht, dinv_t, bl, Xt, NTn, elu);

        // ---- consensus + bilinear ----
        consensus<<<NMAPn, 128, 0, stream>>>(Xt, Xq, u_map, v_map, NMAPn);
        bil_qW<<<DBIL, 128, 0, stream>>>(Xq, u_ptr, W_bil + (long long)i * DBIL * D * D, Tmat);
        bil_out<<<KVn, DBIL, 0, stream>>>(Xt, v_li, Tmat, b_bil + i * DBIL, featb, i);
    }

    // ---- MLP head + output assembly ----
    mlp_head<<<KVn, 128, 0, stream>>>(featb, W1, b1, W2, b2, W3, b3, out);
    hipMemcpyAsync(out_Xq, Xq, (size_t)NQn * D * sizeof(float), hipMemcpyDeviceToDevice, stream);
    hipMemcpyAsync(out_Xt, Xt, (size_t)NTn * D * sizeof(float), hipMemcpyDeviceToDevice, stream);
}